// FrequencyTransformBlock_89043261981406
// MI455X (gfx1250) — compile-verified
//
#include <hip/hip_runtime.h>
#include <math.h>

// ---------------------------------------------------------------------------
// FrequencyTransformBlock for MI455X (gfx1250), f32 end-to-end.
//
// Math:
//   cf_in[b,c] = mean(real(fft2(x)))  ==  x[b,c,0,0]            (DFT identity)
//   cf = sigmoid(cf_in @ W^T + bias)                             (WMMA f32 GEMM)
//   m[b,h,w]  = mean_c x[b,c,h,w]                                (streaming pass 1)
//   sf = C m C - S m S,  C/S = cos/sin(2*pi*k*p/128) tables      (WMMA f32 GEMMs)
//   out = x * (cf[b,c] + sf[b,h,w])                              (streaming pass 2)
//
// Traffic ~805 MB => ~35 us at 23.3 TB/s; GEMM FLOPs negligible, so the
// elementwise passes use float4 (global_load/store_b128) coalesced accesses
// and all matrix work uses V_WMMA_F32_16X16X4_F32 for exact f32 results.
// ---------------------------------------------------------------------------

#define BB  16
#define CC  256
#define HH  128
#define WW  128
#define HWW (HH * WW)            // 16384

typedef __attribute__((ext_vector_type(2))) float v2f;
typedef __attribute__((ext_vector_type(8))) float v8f;

// Persistent device scratch (independent of ws_size); rewritten fully each call.
__device__ float g_m [BB * HWW];     // channel-mean maps        [16][128][128]
__device__ float g_P [BB * HWW];     // P = C * m                [16][128][128]
__device__ float g_Q [BB * HWW];     // Q = S * m                [16][128][128]
__device__ float g_cf[BB * CC];      // sigmoid gate             [16][256]
__device__ float g_sf[BB * HWW];     // sf = P*C - Q*S           [16][128][128]

__device__ __forceinline__ v8f wmma_f32x4(v2f a, v2f b, v8f c) {
    // D = A(16x4) * B(4x16) + C(16x16), all f32, wave32.
    return __builtin_amdgcn_wmma_f32_16x16x4_f32(
        /*neg_a=*/false, a, /*neg_b=*/false, b,
        /*c_mod=*/(short)0, c, /*reuse_a=*/false, /*reuse_b=*/false);
}

// ---------------------------------------------------------------------------
// Kernel 1: m[b,h,w] = (1/256) * sum_c x[b,c,h,w].  65536 threads, float4.
// ---------------------------------------------------------------------------
__global__ void ftb_mean_kernel(const float* __restrict__ x) {
    const int i  = blockIdx.x * blockDim.x + threadIdx.x;   // 0..65535
    const int b  = i >> 12;                                 // 4096 float4 groups / batch
    const int hw = (i & 4095) << 2;
    const float* xp = x + ((size_t)b * CC) * HWW + hw;
    float4 acc = make_float4(0.f, 0.f, 0.f, 0.f);
    for (int c = 0; c < CC; ++c) {
        float4 v = *(const float4*)(xp + (size_t)c * HWW);
        acc.x += v.x; acc.y += v.y; acc.z += v.z; acc.w += v.w;
    }
    const float s = 1.0f / (float)CC;
    float4 o = make_float4(acc.x * s, acc.y * s, acc.z * s, acc.w * s);
    *(float4*)(g_m + (size_t)b * HWW + hw) = o;
}

// ---------------------------------------------------------------------------
// Kernel 2: cf = sigmoid(cf_in @ W^T + bias), cf_in[b,k] = x[b,k,0,0].
// One block (8 waves). M=16(batch) N=256 K=256 via f32 WMMA.
// ---------------------------------------------------------------------------
__global__ void ftb_cf_kernel(const float* __restrict__ x,
                              const float* __restrict__ Wm,
                              const float* __restrict__ bias) {
    __shared__ float sA[BB * CC];                           // 16 KB: cf_in[b][k]
    const int tid = threadIdx.x;
    for (int i = tid; i < BB * CC; i += 256) {
        const int b = i >> 8, k = i & 255;
        sA[i] = x[((size_t)(b * CC + k)) * HWW];            // x[b,k,0,0]
    }
    __syncthreads();

    const int wave = tid >> 5, lane = tid & 31;
    const int half = lane >> 4, lr = lane & 15;

    for (int t = wave; t < 16; t += 8) {                    // N-tile of 16
        v8f acc = {};
        const int n = t * 16 + lr;                          // output channel (B col)
        for (int k0 = 0; k0 < CC; k0 += 4) {
            const int ka = k0 + 2 * half;
            v2f a, bf;
            a.x  = sA[lr * CC + ka];                        // A[row=batch lr][ka]
            a.y  = sA[lr * CC + ka + 1];
            bf.x = Wm[n * CC + ka];                         // B[k][n] = W[n][k]
            bf.y = Wm[n * CC + ka + 1];
            acc = wmma_f32x4(a, bf, acc);
        }
        const float bv = bias[n];
        for (int j = 0; j < 8; ++j) {
            const int row = j + 8 * half;                   // batch
            const float v = acc[j] + bv;
            g_cf[row * CC + n] = 1.0f / (1.0f + expf(-v));
        }
    }
}

// ---------------------------------------------------------------------------
// Kernel 3a: per batch, P = C*m, Q = S*m (128x128x128 f32 WMMA GEMMs).
// 16 blocks x 8 waves; wave w owns output row-block w.
// ---------------------------------------------------------------------------
__global__ void ftb_sf_stage1_kernel() {
    __shared__ float s_ct[128];
    __shared__ float s_st[128];
    const int tid = threadIdx.x;
    if (tid < 128) {
        const float th = (float)tid * (6.28318530717958647692f / 128.0f);
        s_ct[tid] = cosf(th);
        s_st[tid] = sinf(th);
    }
    __syncthreads();

    const int b    = blockIdx.x;
    const int wave = tid >> 5, lane = tid & 31;
    const int half = lane >> 4, lr = lane & 15;
    const float* mb = g_m + (size_t)b * HWW;
    float* Pb = g_P + (size_t)b * HWW;
    float* Qb = g_Q + (size_t)b * HWW;

    const int ti   = wave;                                  // k row-block
    const int krow = ti * 16 + lr;                          // A row (k)
    for (int tj = 0; tj < 8; ++tj) {                        // q col-block
        v8f accP = {}, accQ = {};
        const int qcol = tj * 16 + lr;                      // B col (q)
        for (int p0 = 0; p0 < 128; p0 += 4) {
            const int pa = p0 + 2 * half;
            v2f ac, as_, bm;
            ac.x  = s_ct[(krow * pa) & 127];                // C[k][p]
            ac.y  = s_ct[(krow * (pa + 1)) & 127];
            as_.x = s_st[(krow * pa) & 127];                // S[k][p]
            as_.y = s_st[(krow * (pa + 1)) & 127];
            bm.x  = mb[pa * 128 + qcol];                    // m[p][q]
            bm.y  = mb[(pa + 1) * 128 + qcol];
            accP = wmma_f32x4(ac,  bm, accP);
            accQ = wmma_f32x4(as_, bm, accQ);
        }
        for (int j = 0; j < 8; ++j) {
            const int row = ti * 16 + j + 8 * half;
            const int col = tj * 16 + lr;
            Pb[row * 128 + col] = accP[j];
            Qb[row * 128 + col] = accQ[j];
        }
    }
}

// ---------------------------------------------------------------------------
// Kernel 3b: sf = P*C - Q*S.
// ---------------------------------------------------------------------------
__global__ void ftb_sf_stage2_kernel() {
    __shared__ float s_ct[128];
    __shared__ float s_st[128];
    const int tid = threadIdx.x;
    if (tid < 128) {
        const float th = (float)tid * (6.28318530717958647692f / 128.0f);
        s_ct[tid] = cosf(th);
        s_st[tid] = sinf(th);
    }
    __syncthreads();

    const int b    = blockIdx.x;
    const int wave = tid >> 5, lane = tid & 31;
    const int half = lane >> 4, lr = lane & 15;
    const float* Pb = g_P + (size_t)b * HWW;
    const float* Qb = g_Q + (size_t)b * HWW;
    float* sfb = g_sf + (size_t)b * HWW;

    const int ti   = wave;
    const int krow = ti * 16 + lr;
    for (int tj = 0; tj < 8; ++tj) {
        v8f acc1 = {}, acc2 = {};
        const int lcol = tj * 16 + lr;
        for (int q0 = 0; q0 < 128; q0 += 4) {
            const int qa = q0 + 2 * half;
            v2f ap, aq, bc, bs;
            ap.x = Pb[krow * 128 + qa];                     // P[k][q]
            ap.y = Pb[krow * 128 + qa + 1];
            aq.x = Qb[krow * 128 + qa];                     // Q[k][q]
            aq.y = Qb[krow * 128 + qa + 1];
            bc.x = s_ct[(qa * lcol) & 127];                 // C[q][l]
            bc.y = s_ct[((qa + 1) * lcol) & 127];
            bs.x = s_st[(qa * lcol) & 127];                 // S[q][l]
            bs.y = s_st[((qa + 1) * lcol) & 127];
            acc1 = wmma_f32x4(ap, bc, acc1);
            acc2 = wmma_f32x4(aq, bs, acc2);
        }
        for (int j = 0; j < 8; ++j) {
            const int row = ti * 16 + j + 8 * half;
            const int col = tj * 16 + lr;
            sfb[row * 128 + col] = acc1[j] - acc2[j];       // C m C - S m S
        }
    }
}

// ---------------------------------------------------------------------------
// Kernel 4: out = x * (cf[b,c] + sf[b,h,w]).  float4 streaming.
// ---------------------------------------------------------------------------
__global__ void ftb_apply_kernel(const float* __restrict__ x,
                                 float* __restrict__ out) {
    const size_t i = (size_t)blockIdx.x * blockDim.x + threadIdx.x;
    const size_t e = i << 2;                                // element index
    const int hw = (int)(e & (HWW - 1));
    const int c  = (int)((e >> 14) & 255);
    const int b  = (int)(e >> 22);
    const float cfv = g_cf[b * CC + c];
    const float4 s4 = *(const float4*)(g_sf + (size_t)b * HWW + hw);
    const float4 x4 = *(const float4*)(x + e);
    float4 o;
    o.x = x4.x * (cfv + s4.x);
    o.y = x4.y * (cfv + s4.y);
    o.z = x4.z * (cfv + s4.z);
    o.w = x4.w * (cfv + s4.w);
    *(float4*)(out + e) = o;
}

// ---------------------------------------------------------------------------
extern "C" void kernel_launch(void* const* d_in, const int* in_sizes, int n_in,
                              void* d_out, int out_size, void* d_ws, size_t ws_size,
                              hipStream_t stream) {
    (void)in_sizes; (void)n_in; (void)out_size; (void)d_ws; (void)ws_size;
    const float* x    = (const float*)d_in[0];  // [16,256,128,128]
    const float* Wm   = (const float*)d_in[1];  // [256,256]
    const float* bias = (const float*)d_in[2];  // [256]
    float* out = (float*)d_out;                 // [16,256,128,128]

    ftb_mean_kernel <<<256,   256, 0, stream>>>(x);         // pass 1 over x
    ftb_cf_kernel   <<<1,     256, 0, stream>>>(x, Wm, bias);
    ftb_sf_stage1_kernel<<<16, 256, 0, stream>>>();         // P = C*m, Q = S*m
    ftb_sf_stage2_kernel<<<16, 256, 0, stream>>>();         // sf = P*C - Q*S
    ftb_apply_kernel<<<65536, 256, 0, stream>>>(x, out);    // pass 2 over x
}